// Autoencoder_80161269612703
// MI455X (gfx1250) — compile-verified
//
#include <hip/hip_runtime.h>
#include <math.h>

// ---------------------------------------------------------------------------
// CDNA5 (gfx1250) implementation of the dense sparse-conv autoencoder.
// All pointwise-GEMM / conv / deconv matmuls run on V_WMMA_F32_16X16X4_F32
// (full fp32 operands -> matches reference numerics; bf16/f16 WMMA would
// drift over ~20 chained GEMM layers). One wave32 computes one 16x16 C tile.
// ---------------------------------------------------------------------------

typedef float v2f __attribute__((ext_vector_type(2)));
typedef float v8f __attribute__((ext_vector_type(8)));

__device__ __forceinline__ v8f wmma4(v2f a, v2f b, v8f c) {
  // 8-arg form: (neg_a, A, neg_b, B, c_mod, C, reuse_a, reuse_b)
  return __builtin_amdgcn_wmma_f32_16x16x4_f32(false, a, false, b, (short)0, c,
                                               false, false);
}

// ---------------------------------------------------------------- elementwise
__global__ void zero_k(float* __restrict__ p, size_t n) {
  size_t i = (size_t)blockIdx.x * blockDim.x + threadIdx.x;
  if (i < n) p[i] = 0.f;
}

__global__ void scatter_k(const int* __restrict__ coords,
                          const float* __restrict__ feats,
                          float* __restrict__ G, float* __restrict__ M, int n) {
  int i = blockIdx.x * blockDim.x + threadIdx.x;
  if (i >= n) return;
  int x = coords[3 * i + 0], y = coords[3 * i + 1], z = coords[3 * i + 2];
  size_t f = ((size_t)x * 128 + y) * 128 + z;
  atomicAdd(&G[f], feats[i]);
  M[f] = 1.0f;  // racy but all writers store the same value
}

// level-0 resnet blocks: c==1 so the "GEMMs" are scalars; both reps fused.
__global__ void res0_k(float* __restrict__ G, const float* wn0,
                       const float* w10, const float* w20, const float* wn1,
                       const float* w11, const float* w21, size_t n) {
  size_t i = (size_t)blockIdx.x * blockDim.x + threadIdx.x;
  if (i >= n) return;
  float x = G[i];
  x = x * wn0[0] + fmaxf(fmaxf(x, 0.f) * w10[0], 0.f) * w20[0];
  x = x * wn1[0] + fmaxf(fmaxf(x, 0.f) * w11[0], 0.f) * w21[0];
  G[i] = x;
}

// level-0 downsample conv: c_in = 1 -> 32 (K too small for WMMA).
__global__ void conv0_k(const float* __restrict__ X, const float* __restrict__ Wc,
                        float* __restrict__ O) {
  int t = blockIdx.x * blockDim.x + threadIdx.x;  // voxel*32 + e
  int e = t & 31, v = t >> 5;
  const int Dc = 64, Df = 128;
  int cz = v % Dc, cy = (v / Dc) % Dc, cx = v / (Dc * Dc);
  float acc = 0.f;
#pragma unroll
  for (int p = 0; p < 8; ++p) {
    int i = p >> 2, j = (p >> 1) & 1, k = p & 1;
    size_t f = ((size_t)(2 * cx + i) * Df + (2 * cy + j)) * Df + (2 * cz + k);
    acc += fmaxf(X[f], 0.f) * Wc[p * 32 + e];
  }
  O[(size_t)v * 32 + e] = acc;
}

__global__ void maskpool_k(const float* __restrict__ Mf, float* __restrict__ Mc,
                           int Dc) {
  int v = blockIdx.x * blockDim.x + threadIdx.x;
  if (v >= Dc * Dc * Dc) return;
  int Df = Dc * 2;
  int cz = v % Dc, cy = (v / Dc) % Dc, cx = v / (Dc * Dc);
  float m = 0.f;
#pragma unroll
  for (int p = 0; p < 8; ++p) {
    int i = p >> 2, j = (p >> 1) & 1, k = p & 1;
    size_t f = ((size_t)(2 * cx + i) * Df + (2 * cy + j)) * Df + (2 * cz + k);
    m = fmaxf(m, Mf[f]);
  }
  Mc[v] = m;
}

// ------------------------------------------------------------- WMMA GEMMs
// Fragment loads follow the ISA VGPR layouts:
//  A 16x4 : lane(0-15)=row, vgpr{0,1}=K{0,1}; lanes 16-31 carry K{2,3}.
//  B 4x16 : lane(0-15)=col for K{0,1}; lanes 16-31 same cols for K{2,3}.
//  C/D    : vgpr r -> row r (lanes 0-15) / row r+8 (lanes 16-31), col=lane&15.

// T = relu(X) @ W   (X: rows x C, W: C x C row-major)
__global__ void gemm_relu_k(const float* __restrict__ X,
                            const float* __restrict__ W, float* __restrict__ O,
                            int C) {
  int lane = threadIdx.x & 31, wave = threadIdx.x >> 5;
  int tm = blockIdx.x * 4 + wave, tn = blockIdx.y;
  int r = lane & 15;
  int koff = (lane >> 4) << 1;  // 0 or 2
  size_t row = (size_t)tm * 16 + r;
  int col = tn * 16 + r;
  const float* xr = X + row * C;
  v8f acc = {};
  for (int k0 = 0; k0 < C; k0 += 4) {
    v2f a;
    a.x = fmaxf(xr[k0 + koff], 0.f);
    a.y = fmaxf(xr[k0 + koff + 1], 0.f);
    v2f b;
    b.x = W[(size_t)(k0 + koff) * C + col];
    b.y = W[(size_t)(k0 + koff + 1) * C + col];
    acc = wmma4(a, b, acc);
  }
  int rbase = (lane >> 4) << 3;
#pragma unroll
  for (int q = 0; q < 8; ++q)
    O[((size_t)tm * 16 + rbase + q) * C + col] = acc[q];
}

// O = X @ Wn + relu(T) @ W2   (the AddTable of a resnet block)
__global__ void gemm_fused_k(const float* __restrict__ X,
                             const float* __restrict__ Wn,
                             const float* __restrict__ T,
                             const float* __restrict__ W2,
                             float* __restrict__ O, int C) {
  int lane = threadIdx.x & 31, wave = threadIdx.x >> 5;
  int tm = blockIdx.x * 4 + wave, tn = blockIdx.y;
  int r = lane & 15;
  int koff = (lane >> 4) << 1;
  size_t row = (size_t)tm * 16 + r;
  int col = tn * 16 + r;
  const float* xr = X + row * C;
  const float* tr = T + row * C;
  v8f acc = {};
  for (int k0 = 0; k0 < C; k0 += 4) {
    v2f a;
    a.x = xr[k0 + koff];
    a.y = xr[k0 + koff + 1];
    v2f b;
    b.x = Wn[(size_t)(k0 + koff) * C + col];
    b.y = Wn[(size_t)(k0 + koff + 1) * C + col];
    acc = wmma4(a, b, acc);
  }
  for (int k0 = 0; k0 < C; k0 += 4) {
    v2f a;
    a.x = fmaxf(tr[k0 + koff], 0.f);
    a.y = fmaxf(tr[k0 + koff + 1], 0.f);
    v2f b;
    b.x = W2[(size_t)(k0 + koff) * C + col];
    b.y = W2[(size_t)(k0 + koff + 1) * C + col];
    acc = wmma4(a, b, acc);
  }
  int rbase = (lane >> 4) << 3;
#pragma unroll
  for (int q = 0; q < 8; ++q)
    O[((size_t)tm * 16 + rbase + q) * C + col] = acc[q];
}

// Encoder stride-2 conv: O[coarse, e] = sum_{ijk,d} relu(X[fine(ijk),d]) W[ijk,d,e]
// 8 taps accumulate into the same tile -> looped inside the kernel.
__global__ void conv_down_k(const float* __restrict__ X,
                            const float* __restrict__ Wc, float* __restrict__ O,
                            int Df, int Cin, int Cout) {
  int lane = threadIdx.x & 31, wave = threadIdx.x >> 5;
  int tm = blockIdx.x * 4 + wave, tn = blockIdx.y;
  int r = lane & 15;
  int koff = (lane >> 4) << 1;
  int row = tm * 16 + r;  // coarse voxel id
  int Dc = Df >> 1;
  int cz = row % Dc, cy = (row / Dc) % Dc, cx = row / (Dc * Dc);
  int col = tn * 16 + r;
  v8f acc = {};
  for (int p = 0; p < 8; ++p) {
    int i = p >> 2, j = (p >> 1) & 1, k = p & 1;
    size_t f = ((size_t)(2 * cx + i) * Df + (2 * cy + j)) * Df + (2 * cz + k);
    const float* ap = X + f * Cin;
    const float* wp = Wc + (size_t)p * Cin * Cout;
    for (int k0 = 0; k0 < Cin; k0 += 4) {
      v2f a;
      a.x = fmaxf(ap[k0 + koff], 0.f);
      a.y = fmaxf(ap[k0 + koff + 1], 0.f);
      v2f b;
      b.x = wp[(size_t)(k0 + koff) * Cout + col];
      b.y = wp[(size_t)(k0 + koff + 1) * Cout + col];
      acc = wmma4(a, b, acc);
    }
  }
  int rbase = (lane >> 4) << 3;
#pragma unroll
  for (int q = 0; q < 8; ++q) {
    int m = tm * 16 + rbase + q;
    O[(size_t)m * Cout + col] = acc[q];
  }
}

// Decoder transposed conv: each (i,j,k) tap writes disjoint fine voxels ->
// taps on gridDim.z; mask multiply fused into the scattered store.
__global__ void deconv_k(const float* __restrict__ X,
                         const float* __restrict__ Wd,
                         const float* __restrict__ Mfine, float* __restrict__ O,
                         int Dc, int Cin, int Cout) {
  int lane = threadIdx.x & 31, wave = threadIdx.x >> 5;
  int tm = blockIdx.x * 4 + wave, tn = blockIdx.y, p = blockIdx.z;
  int i = p >> 2, j = (p >> 1) & 1, k = p & 1;
  int r = lane & 15;
  int koff = (lane >> 4) << 1;
  size_t row = (size_t)tm * 16 + r;  // coarse voxel id
  int col = tn * 16 + r;
  const float* ap = X + row * Cin;
  const float* wp = Wd + (size_t)p * Cin * Cout;
  v8f acc = {};
  for (int k0 = 0; k0 < Cin; k0 += 4) {
    v2f a;
    a.x = fmaxf(ap[k0 + koff], 0.f);
    a.y = fmaxf(ap[k0 + koff + 1], 0.f);
    v2f b;
    b.x = wp[(size_t)(k0 + koff) * Cout + col];
    b.y = wp[(size_t)(k0 + koff + 1) * Cout + col];
    acc = wmma4(a, b, acc);
  }
  int Df = Dc * 2;
  int rbase = (lane >> 4) << 3;
#pragma unroll
  for (int q = 0; q < 8; ++q) {
    int m = tm * 16 + rbase + q;
    int cz = m % Dc, cy = (m / Dc) % Dc, cx = m / (Dc * Dc);
    size_t f = ((size_t)(2 * cx + i) * Df + (2 * cy + j)) * Df + (2 * cz + k);
    O[f * Cout + col] = acc[q] * Mfine[f];
  }
}

// Final decoder level (c_out=1) + sigmoid*mask0, one thread per fine voxel.
__global__ void final_k(const float* __restrict__ X, const float* __restrict__ Wd,
                        const float* __restrict__ M0, float* __restrict__ Out) {
  const int Df = 128, Dc = 64, Cin = 32;
  size_t f = (size_t)blockIdx.x * blockDim.x + threadIdx.x;
  int z = (int)(f % Df), y = (int)((f / Df) % Df), x = (int)(f / (Df * Df));
  int i = x & 1, j = y & 1, k = z & 1;
  size_t ci = (((size_t)(x >> 1) * Dc + (y >> 1)) * Dc + (z >> 1)) * Cin;
  const float* wp = Wd + (size_t)(((i * 2 + j) * 2 + k)) * Cin;  // c_out == 1
  float acc = 0.f;
#pragma unroll
  for (int d = 0; d < Cin; ++d) acc += fmaxf(X[ci + d], 0.f) * wp[d];
  float m = M0[f];
  float v = acc * m;
  Out[f] = (1.0f / (1.0f + expf(-v))) * m;
}

// ---------------------------------------------------------------------------
extern "C" void kernel_launch(void* const* d_in, const int* in_sizes, int n_in,
                              void* d_out, int out_size, void* d_ws,
                              size_t ws_size, hipStream_t stream) {
  const int* coords = (const int*)d_in[0];
  const float* feats = (const float*)d_in[1];
  const float* W[32];
  for (int i = 0; i < 32; ++i) W[i] = (const float*)d_in[2 + i];
  // enc level l: W[7l+0..2]=rb0(wnin,w1,w2), W[7l+3..5]=rb1, W[7l+6]=wconv.
  // dec level l: W[28+l].
  float* out = (float*)d_out;

  // workspace layout (floats); total ~37.7M floats (~151 MB)
  float* ws = (float*)d_ws;
  size_t off = 0;
  float* g0 = ws + off; off += 2097152;   // 128^3 x 1
  float* m0 = ws + off; off += 2097152;   // 128^3 mask (contiguous after g0)
  float* x1a = ws + off; off += 8388608;  // 64^3 x 32
  float* x1b = ws + off; off += 8388608;
  float* x1t = ws + off; off += 8388608;
  float* m1 = ws + off; off += 262144;
  float* x2a = ws + off; off += 2097152;  // 32^3 x 64
  float* x2b = ws + off; off += 2097152;
  float* x2t = ws + off; off += 2097152;
  float* m2 = ws + off; off += 32768;
  float* x3a = ws + off; off += 524288;   // 16^3 x 128
  float* x3b = ws + off; off += 524288;
  float* x3t = ws + off; off += 524288;
  float* m3 = ws + off; off += 4096;
  float* x4 = ws + off; off += 131072;    // 8^3 x 256
  (void)ws_size; (void)in_sizes; (void)n_in; (void)out_size;

  // input layer: zero grid+mask (contiguous), scatter points
  zero_k<<<4194304 / 256, 256, 0, stream>>>(g0, (size_t)4194304);
  scatter_k<<<(200000 + 255) / 256, 256, 0, stream>>>(coords, feats, g0, m0,
                                                      200000);
  // ---- level 0 (c=1): scalar resblocks + 1->32 downsample conv
  res0_k<<<8192, 256, 0, stream>>>(g0, W[0], W[1], W[2], W[3], W[4], W[5],
                                   (size_t)2097152);
  conv0_k<<<32768, 256, 0, stream>>>(g0, W[6], x1a);
  maskpool_k<<<1024, 256, 0, stream>>>(m0, m1, 64);

  // ---- level 1 (64^3, C=32): rows=262144 -> 16384 tiles -> grid.x=4096
  gemm_relu_k<<<dim3(4096, 2), 128, 0, stream>>>(x1a, W[8], x1t, 32);
  gemm_fused_k<<<dim3(4096, 2), 128, 0, stream>>>(x1a, W[7], x1t, W[9], x1b, 32);
  gemm_relu_k<<<dim3(4096, 2), 128, 0, stream>>>(x1b, W[11], x1t, 32);
  gemm_fused_k<<<dim3(4096, 2), 128, 0, stream>>>(x1b, W[10], x1t, W[12], x1a, 32);
  conv_down_k<<<dim3(512, 4), 128, 0, stream>>>(x1a, W[13], x2a, 64, 32, 64);
  maskpool_k<<<128, 256, 0, stream>>>(m1, m2, 32);

  // ---- level 2 (32^3, C=64): rows=32768 -> 2048 tiles -> grid.x=512
  gemm_relu_k<<<dim3(512, 4), 128, 0, stream>>>(x2a, W[15], x2t, 64);
  gemm_fused_k<<<dim3(512, 4), 128, 0, stream>>>(x2a, W[14], x2t, W[16], x2b, 64);
  gemm_relu_k<<<dim3(512, 4), 128, 0, stream>>>(x2b, W[18], x2t, 64);
  gemm_fused_k<<<dim3(512, 4), 128, 0, stream>>>(x2b, W[17], x2t, W[19], x2a, 64);
  conv_down_k<<<dim3(64, 8), 128, 0, stream>>>(x2a, W[20], x3a, 32, 64, 128);
  maskpool_k<<<16, 256, 0, stream>>>(m2, m3, 16);

  // ---- level 3 (16^3, C=128): rows=4096 -> 256 tiles -> grid.x=64
  gemm_relu_k<<<dim3(64, 8), 128, 0, stream>>>(x3a, W[22], x3t, 128);
  gemm_fused_k<<<dim3(64, 8), 128, 0, stream>>>(x3a, W[21], x3t, W[23], x3b, 128);
  gemm_relu_k<<<dim3(64, 8), 128, 0, stream>>>(x3b, W[25], x3t, 128);
  gemm_fused_k<<<dim3(64, 8), 128, 0, stream>>>(x3b, W[24], x3t, W[26], x3a, 128);
  conv_down_k<<<dim3(8, 16), 128, 0, stream>>>(x3a, W[27], x4, 16, 128, 256);

  // ---- decoder (8 taps in gridDim.z; mask fused into store)
  deconv_k<<<dim3(8, 8, 8), 128, 0, stream>>>(x4, W[28], m3, x3b, 8, 256, 128);
  deconv_k<<<dim3(64, 4, 8), 128, 0, stream>>>(x3b, W[29], m2, x2b, 16, 128, 64);
  deconv_k<<<dim3(512, 2, 8), 128, 0, stream>>>(x2b, W[30], m1, x1b, 32, 64, 32);
  final_k<<<8192, 256, 0, stream>>>(x1b, W[31], m0, out);
}